// TrajSparseDGATLayer_19971597926785
// MI455X (gfx1250) — compile-verified
//
#include <hip/hip_runtime.h>
#include <hip/hip_bf16.h>
#include <math.h>
#include <stdint.h>

// Problem dims (fixed by the reference)
#define B_    4
#define N_    10000
#define E_    160000
#define FIN_  128
#define NH_   8
#define FOUT_ 16
#define HID_  128

typedef float v2f __attribute__((ext_vector_type(2)));
typedef float v8f __attribute__((ext_vector_type(8)));

// ---- workspace layout (float offsets) ----
static constexpr size_t OFF_PROJ = 0;                       // B*N*HID  = 5,120,000
static constexpr size_t OFF_AGG  = OFF_PROJ + (size_t)B_*N_*HID_;   // 5,120,000 (init = x@W_skip)
static constexpr size_t OFF_SSRC = OFF_AGG  + (size_t)B_*N_*HID_;   // B*N*NH = 320,000
static constexpr size_t OFF_STRG = OFF_SSRC + (size_t)B_*N_*NH_;    // 320,000
static constexpr size_t OFF_DEN  = OFF_STRG + (size_t)B_*N_*NH_;    // 320,000
static constexpr size_t OFF_EXPS = OFF_DEN  + (size_t)B_*N_*NH_;    // B*E*NH = 5,120,000
static constexpr size_t OFF_MISC = OFF_EXPS + (size_t)B_*E_*NH_;    // [0..7]=c_tp [8..15]=c_ttp [16]=gmax(enc u32)

// monotone float<->uint encoding so unsigned atomicMax orders like float max
__device__ __forceinline__ unsigned enc_f32(float f) {
  unsigned u = __float_as_uint(f);
  return (u & 0x80000000u) ? ~u : (u | 0x80000000u);
}
__device__ __forceinline__ float dec_f32(unsigned e) {
  return (e & 0x80000000u) ? __uint_as_float(e & 0x7FFFFFFFu) : __uint_as_float(~e);
}
__device__ __forceinline__ float elu1(float x) { return x > 0.f ? x : (expf(x) - 1.f); }

// ---------------------------------------------------------------------------
// K0: zero softmax denominators, fold the rank-1 tp/ttp paths into 8 scalars
//     per head, reset the global-max accumulator.
// ---------------------------------------------------------------------------
__global__ void k_init(float* __restrict__ ws,
                       const float* __restrict__ W_tp, const float* __restrict__ W_ttp,
                       const float* __restrict__ sc_tp, const float* __restrict__ sc_ttp) {
  int idx = blockIdx.x * blockDim.x + threadIdx.x;
  if (idx < B_ * N_ * NH_) ws[OFF_DEN + idx] = 0.f;
  if (blockIdx.x == 0) {
    int t = threadIdx.x;
    if (t < NH_) {
      float a = 0.f;
      for (int f = 0; f < FOUT_; ++f) a += W_tp[t * FOUT_ + f] * sc_tp[t * FOUT_ + f];
      ws[OFF_MISC + t] = a;
    } else if (t < 2 * NH_) {
      int h = t - NH_;
      float a = 0.f;
      for (int f = 0; f < FOUT_; ++f) a += W_ttp[h * FOUT_ + f] * sc_ttp[h * FOUT_ + f];
      ws[OFF_MISC + NH_ + h] = a;
    } else if (t == 2 * NH_) {
      *(unsigned*)(ws + OFF_MISC + 16) = 0u;  // encoded -inf sentinel
    }
  }
}

// ---------------------------------------------------------------------------
// K1: fp32 WMMA GEMM.  Block = 256 thr = 8 waves, owns a 16-row A tile.
//     A tile (16x128 f32 = 8 KB) staged into LDS with the CDNA5 async copy
//     engine (GLOBAL_LOAD_ASYNC_TO_LDS_B128, ASYNCcnt), then
//     s_wait_asynccnt 0 + workgroup barrier.
//     The LDS address operand is derived from &sA[...] itself (ptrtoint) and
//     the pointer is passed to the asm, so sA is "captured" and the memory
//     clobber forces the compiler to see the LDS write (round-2 bug fix).
//     Wave w computes output cols [16w,16w+16) for BOTH W_proj (-> proj) and
//     W_skip (-> agg init), sharing A fragments; K marched in steps of 4 via
//     V_WMMA_F32_16X16X4_F32 (exact fp32 MMA path).
// A frag (16x4 f32): lanes 0-15 M=lane K={0,1}; lanes 16-31 M=lane-16 K={2,3}
// B frag (4x16 f32): lanes 0-15 N=lane K={0,1}; lanes 16-31 N=lane-16 K={2,3}
// C/D (16x16 f32):   VGPR g = row (g + 8*(lane>=16)), col = lane%16
// ---------------------------------------------------------------------------
__global__ __launch_bounds__(256) void k_gemm(const float* __restrict__ x,
                                              const float* __restrict__ Wp,
                                              const float* __restrict__ Wsk,
                                              float* __restrict__ ws) {
  __shared__ float sA[16][FIN_];  // 8 KB
  const int rowBase = blockIdx.x * 16;
  {
    int r = threadIdx.x >> 4;         // 0..15
    int c = (threadIdx.x & 15) * 8;   // 0..120 : this thread stages 8 floats
    void* dstp = (void*)&sA[r][c];
    // Low 32 bits of a generic shared address == wave-relative LDS byte addr.
    unsigned lds_off = (unsigned)(uintptr_t)dstp;
    unsigned long long gaddr =
        (unsigned long long)(const void*)(x + (size_t)(rowBase + r) * FIN_ + c);
    // Two 16B async copies straight into LDS (INST_OFFSET applies to both sides).
    asm volatile("global_load_async_to_lds_b128 %0, %1, off"
                 :: "v"(lds_off), "v"(gaddr), "v"(dstp) : "memory");
    asm volatile("global_load_async_to_lds_b128 %0, %1, off offset:16"
                 :: "v"(lds_off), "v"(gaddr), "v"(dstp) : "memory");
    asm volatile("s_wait_asynccnt 0x0" ::: "memory");
  }
  __syncthreads();

  const int lane  = threadIdx.x & 31;
  const int wv    = threadIdx.x >> 5;     // 0..7 -> column tile
  const int cb    = wv * 16;
  const int mr    = lane & 15;            // row (A) / col (B,C)
  const int khalf = (lane >> 4) * 2;      // 0 or 2

  v8f accP = {};
  v8f accS = {};
  for (int k0 = 0; k0 < FIN_; k0 += 4) {
    v2f a;
    a.x = sA[mr][k0 + khalf];
    a.y = sA[mr][k0 + khalf + 1];
    v2f bp, bs;
    bp.x = Wp [(k0 + khalf)     * HID_ + cb + mr];
    bp.y = Wp [(k0 + khalf + 1) * HID_ + cb + mr];
    bs.x = Wsk[(k0 + khalf)     * HID_ + cb + mr];
    bs.y = Wsk[(k0 + khalf + 1) * HID_ + cb + mr];
    accP = __builtin_amdgcn_wmma_f32_16x16x4_f32(false, a, false, bp, (short)0, accP, false, false);
    accS = __builtin_amdgcn_wmma_f32_16x16x4_f32(false, a, false, bs, (short)0, accS, false, false);
  }

  float* proj = ws + OFF_PROJ;
  float* agg  = ws + OFF_AGG;
  const int rowOff = (lane >> 4) * 8;
  for (int g = 0; g < 8; ++g) {
    size_t o = (size_t)(rowBase + rowOff + g) * HID_ + cb + mr;
    proj[o] = accP[g];
    agg[o]  = accS[g];   // skip-connection initializes the aggregate
  }
}

// ---------------------------------------------------------------------------
// K2: per-(b,n,h) source/target attention scores (dot of 16-wide head slice).
// ---------------------------------------------------------------------------
__global__ void k_nodescore(float* __restrict__ ws,
                            const float* __restrict__ sc_src,
                            const float* __restrict__ sc_trg) {
  int idx = blockIdx.x * blockDim.x + threadIdx.x;
  if (idx >= B_ * N_ * NH_) return;
  int h  = idx & (NH_ - 1);
  int bn = idx >> 3;
  const float* p = ws + OFF_PROJ + (size_t)bn * HID_ + h * FOUT_;
  const float* a = sc_src + h * FOUT_;
  const float* t = sc_trg + h * FOUT_;
  float s0 = 0.f, s1 = 0.f;
#pragma unroll
  for (int f = 0; f < FOUT_; ++f) { float v = p[f]; s0 += v * a[f]; s1 += v * t[f]; }
  ws[OFF_SSRC + idx] = s0;
  ws[OFF_STRG + idx] = s1;
}

// ---------------------------------------------------------------------------
// K3: global max over leaky-relu'd scores (wave-reduced, one atomic/wave).
// ---------------------------------------------------------------------------
__global__ void k_edgemax(float* __restrict__ ws, const int* __restrict__ ei,
                          const float* __restrict__ ep, const float* __restrict__ tep) {
  int idx = blockIdx.x * blockDim.x + threadIdx.x;  // b*E + e
  float m = -3.4e38f;
  if (idx < B_ * E_) {
    int b = idx / E_, e = idx - b * E_;
    int s = ei[e], t = ei[E_ + e];
    const float* ss   = ws + OFF_SSRC + (size_t)(b * N_ + s) * NH_;
    const float* st   = ws + OFF_STRG + (size_t)(b * N_ + t) * NH_;
    const float* ctp  = ws + OFF_MISC;
    const float* cttp = ws + OFF_MISC + NH_;
    float cep = ep[e], ctep = tep[idx];
#pragma unroll
    for (int h = 0; h < NH_; ++h) {
      float sc = ss[h] + st[h] + cep * ctp[h] + ctep * cttp[h];
      sc = sc > 0.f ? sc : 0.2f * sc;
      m = fmaxf(m, sc);
    }
  }
#pragma unroll
  for (int off = 16; off >= 1; off >>= 1) m = fmaxf(m, __shfl_xor(m, off, 32));
  if ((threadIdx.x & 31) == 0) atomicMax((unsigned*)(ws + OFF_MISC + 16), enc_f32(m));
}

// ---------------------------------------------------------------------------
// K4: exp(score - gmax), store per-edge, scatter-add softmax denominators.
// ---------------------------------------------------------------------------
__global__ void k_edgeexp(float* __restrict__ ws, const int* __restrict__ ei,
                          const float* __restrict__ ep, const float* __restrict__ tep) {
  int idx = blockIdx.x * blockDim.x + threadIdx.x;
  if (idx >= B_ * E_) return;
  int b = idx / E_, e = idx - b * E_;
  int s = ei[e], t = ei[E_ + e];
  const float gmax = dec_f32(*(const unsigned*)(ws + OFF_MISC + 16));
  const float* ss   = ws + OFF_SSRC + (size_t)(b * N_ + s) * NH_;
  const float* st   = ws + OFF_STRG + (size_t)(b * N_ + t) * NH_;
  const float* ctp  = ws + OFF_MISC;
  const float* cttp = ws + OFF_MISC + NH_;
  float cep = ep[e], ctep = tep[idx];
  float* exps = ws + OFF_EXPS + (size_t)idx * NH_;
  float* den  = ws + OFF_DEN  + (size_t)(b * N_ + t) * NH_;
#pragma unroll
  for (int h = 0; h < NH_; ++h) {
    float sc = ss[h] + st[h] + cep * ctp[h] + ctep * cttp[h];
    sc = sc > 0.f ? sc : 0.2f * sc;
    float v = expf(sc - gmax);
    exps[h] = v;
    atomicAdd(den + h, v);
  }
}

// ---------------------------------------------------------------------------
// K5: one wave per (b,e): gather proj[b,src] (float4/lane), scale by attn,
//     scatter-add into agg[b,trg].  The dominant (L2-resident) pass.
// ---------------------------------------------------------------------------
__global__ __launch_bounds__(256) void k_agg(float* __restrict__ ws, const int* __restrict__ ei) {
  const int lane = threadIdx.x & 31;
  const long long widx = (long long)blockIdx.x * 8 + (threadIdx.x >> 5);
  if (widx >= (long long)B_ * E_) return;
  int b = (int)(widx / E_), e = (int)(widx - (long long)b * E_);
  int s = ei[e], t = ei[E_ + e];

  const float* prow = ws + OFF_PROJ + (size_t)(b * N_ + s) * HID_;
  __builtin_prefetch(prow + lane * 4, 0, 2);   // global_prefetch hint

  int   h    = lane >> 2;                      // 4 floats/lane -> head = lane/4
  float ex   = ws[OFF_EXPS + (size_t)widx * NH_ + h];
  float dn   = ws[OFF_DEN  + (size_t)(b * N_ + t) * NH_ + h];
  float attn = ex / (dn + 1e-16f);

  float4 p = *(const float4*)(prow + lane * 4);
  float* arow = ws + OFF_AGG + (size_t)(b * N_ + t) * HID_ + lane * 4;
  atomicAdd(arow + 0, p.x * attn);
  atomicAdd(arow + 1, p.y * attn);
  atomicAdd(arow + 2, p.z * attn);
  atomicAdd(arow + 3, p.w * attn);
}

// ---------------------------------------------------------------------------
// K6: out = ELU(agg + bias)   (agg already contains skip + scatter sums)
// ---------------------------------------------------------------------------
__global__ void k_final(const float* __restrict__ ws, const float* __restrict__ bias,
                        float* __restrict__ out) {
  int idx = blockIdx.x * blockDim.x + threadIdx.x;
  if (idx >= (B_ * N_ * HID_) / 4) return;
  float4 v  = *(const float4*)(ws + OFF_AGG + (size_t)idx * 4);
  int    cb = (idx * 4) & (HID_ - 1);
  float4 bb = *(const float4*)(bias + cb);
  float4 r;
  r.x = elu1(v.x + bb.x);
  r.y = elu1(v.y + bb.y);
  r.z = elu1(v.z + bb.z);
  r.w = elu1(v.w + bb.w);
  ((float4*)out)[idx] = r;
}

extern "C" void kernel_launch(void* const* d_in, const int* in_sizes, int n_in,
                              void* d_out, int out_size, void* d_ws, size_t ws_size,
                              hipStream_t stream) {
  const float* x     = (const float*)d_in[0];
  const float* ep    = (const float*)d_in[1];
  const float* tep   = (const float*)d_in[2];
  const float* Wp    = (const float*)d_in[3];
  const float* Wtp   = (const float*)d_in[4];
  const float* Wttp  = (const float*)d_in[5];
  const float* Wsk   = (const float*)d_in[6];
  const float* scsrc = (const float*)d_in[7];
  const float* sctrg = (const float*)d_in[8];
  const float* sctp  = (const float*)d_in[9];
  const float* scttp = (const float*)d_in[10];
  const float* bias  = (const float*)d_in[11];
  const int*   ei    = (const int*)d_in[12];   // harness contract: integer -> const int*
  float* ws  = (float*)d_ws;
  float* out = (float*)d_out;

  k_init     <<<(B_ * N_ * NH_ + 255) / 256, 256, 0, stream>>>(ws, Wtp, Wttp, sctp, scttp);
  k_gemm     <<<(B_ * N_) / 16,              256, 0, stream>>>(x, Wp, Wsk, ws);
  k_nodescore<<<(B_ * N_ * NH_ + 255) / 256, 256, 0, stream>>>(ws, scsrc, sctrg);
  k_edgemax  <<<(B_ * E_ + 255) / 256,       256, 0, stream>>>(ws, ei, ep, tep);
  k_edgeexp  <<<(B_ * E_ + 255) / 256,       256, 0, stream>>>(ws, ei, ep, tep);
  k_agg      <<<(B_ * E_ + 7) / 8,           256, 0, stream>>>(ws, ei);
  k_final    <<<(B_ * N_ * HID_ / 4 + 255) / 256, 256, 0, stream>>>(ws, bias, out);
}